// WeightNet_57509612093536
// MI455X (gfx1250) — compile-verified
//
#include <hip/hip_runtime.h>
#include <hip/hip_bf16.h>
#include <stdint.h>

// WeightNet gather-matvec:  y[n,c] = leakyrelu( sum_k pos[n,k]*W[idx[n],k,c] + b[idx[n],c] )
// N=500000, FANIN=7, C=64, L=2048.
// Roofline: 0.45 GFLOP vs ~145 MB min HBM traffic (128 MB output write dominates)
// => pure bandwidth kernel, ~6 us at 23.3 TB/s. W (3.67 MB) is kept L2-resident;
// streaming data (pos/idx in, out) uses CDNA5 non-temporal hints so it cannot
// evict the reused W/bias rows.

#define FANIN 7
#define CCH   64
#define SPB   16      // samples per block (16 threads per sample, float4 per lane)
#define TPB   256     // 8 wave32s

typedef float v4f __attribute__((ext_vector_type(4)));

// gfx1250 async global->LDS copy (ASYNCcnt path), non-temporal: pos/idx are
// read-once streams. VDST VGPR = 32-bit workgroup-relative LDS byte address,
// VADDR = 64-bit global address, saddr = off.
__device__ __forceinline__ void async_load_f32_to_lds_nt(const float* g, void* lds_generic) {
    unsigned lds_off = (unsigned)(uintptr_t)lds_generic;
    asm volatile("global_load_async_to_lds_b32 %0, %1, off th:TH_LOAD_NT"
                 :
                 : "v"(lds_off), "v"(g)
                 : "memory");
}

__device__ __forceinline__ void wait_asynccnt0() {
    asm volatile("s_wait_asynccnt 0" ::: "memory");
}

__global__ __launch_bounds__(TPB) void weightnet_kernel(
    const float* __restrict__ pos,   // [N,7]
    const int*   __restrict__ sidx,  // [N]
    const float* __restrict__ W,     // [L,7,64]
    const float* __restrict__ bias,  // [L,64]
    float*       __restrict__ out,   // [N,64]
    int n_total)
{
    __shared__ float s_pos[SPB * 8];   // stride 8 keeps banks clean
    __shared__ int   s_idx[SPB];

    const int t  = threadIdx.x;
    const int nb = blockIdx.x * SPB;

    // Stage this block's pos (16x7 floats) and seq_idx (16 ints) into LDS via
    // the async data-mover path: one 4-byte copy per participating lane.
    if (t < SPB * FANIN) {
        const int s = t / FANIN;
        const int k = t - s * FANIN;
        if (nb + s < n_total)
            async_load_f32_to_lds_nt(pos + (size_t)(nb + s) * FANIN + k, &s_pos[s * 8 + k]);
    } else if (t < SPB * FANIN + SPB) {
        const int s = t - SPB * FANIN;
        if (nb + s < n_total)
            async_load_f32_to_lds_nt((const float*)(sidx + nb + s), &s_idx[s]);
    }
    wait_asynccnt0();
    __syncthreads();

    const int s  = t >> 4;   // sample within block
    const int cq = t & 15;   // channel quad: owns channels [4*cq, 4*cq+3]
    const int n  = nb + s;
    if (n >= n_total) return;

    const int row = s_idx[s];
    const float* wrow = W    + (size_t)row * (FANIN * CCH) + cq * 4;
    const float* brow = bias + (size_t)row * CCH           + cq * 4;

    // Kick the gathered (L2-resident) W row toward the WGP early.
    __builtin_prefetch(wrow, 0, 0);

    v4f acc = *(const v4f*)brow;                 // regular-temporal: reused row
#pragma unroll
    for (int k = 0; k < FANIN; ++k) {
        const float p = s_pos[s * 8 + k];
        const v4f   w = *(const v4f*)(wrow + (size_t)k * CCH);  // regular-temporal
        acc = w * p + acc;                       // contracts to v_fma_f32 x4
    }

    const float ns = 0.2f;
    acc.x = (acc.x >= 0.f) ? acc.x : ns * acc.x;
    acc.y = (acc.y >= 0.f) ? acc.y : ns * acc.y;
    acc.z = (acc.z >= 0.f) ? acc.z : ns * acc.z;
    acc.w = (acc.w >= 0.f) ? acc.w : ns * acc.w;

    // Output is write-once: non-temporal store so 128 MB of result stream
    // does not evict the W table from WGP$/L2.
    __builtin_nontemporal_store(acc, (v4f*)(out + (size_t)n * CCH + cq * 4));
}

extern "C" void kernel_launch(void* const* d_in, const int* in_sizes, int n_in,
                              void* d_out, int out_size, void* d_ws, size_t ws_size,
                              hipStream_t stream) {
    const float* pos  = (const float*)d_in[0];   // [N,7] f32
    const int*   sidx = (const int*)  d_in[1];   // [N]   i32
    const float* W    = (const float*)d_in[2];   // [L,7,64] f32
    const float* bias = (const float*)d_in[3];   // [L,64] f32
    float*       out  = (float*)d_out;           // [N,64] f32

    const int n      = in_sizes[1];              // N samples
    const int blocks = (n + SPB - 1) / SPB;

    weightnet_kernel<<<blocks, TPB, 0, stream>>>(pos, sidx, W, bias, out, n);
}